// ButterworthFilter_4148938408285
// MI455X (gfx1250) — compile-verified
//
#include <hip/hip_runtime.h>

typedef __attribute__((ext_vector_type(2))) float v2f;
typedef __attribute__((ext_vector_type(8))) float v8f;

#define T_LEN 96000
#define GRP   256                 // 16 blocks of 16 samples
#define NGRP  (T_LEN / GRP)       // 375

struct Biquad { float b0, b1, b2, a1, a2; };

__device__ __forceinline__ void step2(const Biquad& q0, const Biquad& q1,
                                      float x, float& w1a, float& w2a,
                                      float& w1b, float& w2b, float& y)
{
    float w0a = x - q0.a1 * w1a - q0.a2 * w2a;
    float ya  = q0.b0 * w0a + q0.b1 * w1a + q0.b2 * w2a;
    w2a = w1a; w1a = w0a;
    float w0b = ya - q1.a1 * w1b - q1.a2 * w2b;
    y   = q1.b0 * w0b + q1.b1 * w1b + q1.b2 * w2b;
    w2b = w1b; w1b = w0b;
}

// v_readlane_b32: reads source lane regardless of EXEC (safe cross-lane gather)
__device__ __forceinline__ float rdlane(float v, int l)
{
    return __int_as_float(__builtin_amdgcn_readlane(__float_as_int(v), l));
}

__global__ __launch_bounds__(32)
void biquad_wmma_kernel(const float* __restrict__ x,
                        const float* __restrict__ sos,
                        float* __restrict__ out)
{
    // One-time setup matrices only (no steady-state LDS traffic)
    __shared__ float lD[256];  // D[m*16+k]  : 16x16 Toeplitz zero-state impulse
    __shared__ float lE[64];   // E[s*16+j]  : block input -> block end-state
    __shared__ float lH[64];   // H[m*4+s]   : block init-state -> block output
    __shared__ float lA[16];   // A^16 [r*4+c]

    const int lane = threadIdx.x;
    const int sig  = blockIdx.x;

    if (lane == 0) {
        Biquad q0, q1;
        {
            float a0 = sos[3];
            q0.b0 = sos[0] / a0;  q0.b1 = sos[1] / a0;  q0.b2 = sos[2] / a0;
            q0.a1 = sos[4] / a0;  q0.a2 = sos[5] / a0;
            a0 = sos[9];
            q1.b0 = sos[6] / a0;  q1.b1 = sos[7] / a0;  q1.b2 = sos[8] / a0;
            q1.a1 = sos[10] / a0; q1.a2 = sos[11] / a0;
        }
        // Impulse run: h[t] (-> D) and states after t+1 steps (-> E columns)
        float h[16];
        float se[4][17];
        {
            float w1a = 0, w2a = 0, w1b = 0, w2b = 0, y;
            for (int t = 0; t < 16; ++t) {
                step2(q0, q1, (t == 0) ? 1.0f : 0.0f, w1a, w2a, w1b, w2b, y);
                h[t] = y;
                se[0][t + 1] = w1a; se[1][t + 1] = w2a;
                se[2][t + 1] = w1b; se[3][t + 1] = w2b;
            }
        }
        for (int i = 0; i < 16; ++i)
            for (int j = 0; j < 16; ++j)
                lD[i * 16 + j] = (i >= j) ? h[i - j] : 0.0f;
        for (int s = 0; s < 4; ++s)
            for (int j = 0; j < 16; ++j)
                lE[s * 16 + j] = se[s][16 - j];   // impulse at j -> state 16-j steps later
        // Homogeneous runs from unit states: H (outputs) and A^16 (end states)
        for (int s = 0; s < 4; ++s) {
            float st[4] = {0, 0, 0, 0}; st[s] = 1.0f;
            float w1a = st[0], w2a = st[1], w1b = st[2], w2b = st[3], y;
            for (int t = 0; t < 16; ++t) {
                step2(q0, q1, 0.0f, w1a, w2a, w1b, w2b, y);
                lH[t * 4 + s] = y;
            }
            lA[0 * 4 + s] = w1a; lA[1 * 4 + s] = w2a;
            lA[2 * 4 + s] = w1b; lA[3 * 4 + s] = w2b;
        }
    }
    __syncthreads();

    const int half = lane >> 4;     // A/B layout: lanes 0-15 -> K 0/1, 16-31 -> K 2/3
    const int m    = lane & 15;

    // Constant A-fragments (16x4 per WMMA K-chunk)
    v2f aD[4], aE[4], aH;
#pragma unroll
    for (int c = 0; c < 4; ++c) {
        int k = 4 * c + 2 * half;
        aD[c][0] = lD[m * 16 + k];
        aD[c][1] = lD[m * 16 + k + 1];
        aE[c][0] = (m < 4) ? lE[m * 16 + k]     : 0.0f;
        aE[c][1] = (m < 4) ? lE[m * 16 + k + 1] : 0.0f;
    }
    aH[0] = lH[m * 4 + 2 * half];
    aH[1] = lH[m * 4 + 2 * half + 1];

    // A^16 into registers (uniform across lanes)
    float A[16];
#pragma unroll
    for (int i = 0; i < 16; ++i) A[i] = lA[i];

    const float* xin  = x   + (size_t)sig * T_LEN;
    float*       yout = out + (size_t)sig * T_LEN;

    // Per-lane global pointer to its B-fragment elements: X[k][n=m] pairs
    const float* px = xin + m * 16 + 2 * half;

    // Group carry state: uniform across lanes, lives in registers
    float s0 = 0.0f, s1 = 0.0f, s2 = 0.0f, s3 = 0.0f;

    // Preload group 0 fragments
    v2f bX[4];
#pragma unroll
    for (int c = 0; c < 4; ++c) bX[c] = *(const v2f*)(px + 4 * c);

    for (int g = 0; g < NGRP; ++g) {
        // Y_local = D @ X ; S_local = E_pad @ X   (4 chained K=4 WMMAs each)
        v8f yv = {0, 0, 0, 0, 0, 0, 0, 0};
        v8f sv = {0, 0, 0, 0, 0, 0, 0, 0};
#pragma unroll
        for (int c = 0; c < 4; ++c) {
            yv = __builtin_amdgcn_wmma_f32_16x16x4_f32(false, aD[c], false, bX[c],
                                                       (short)0, yv, false, false);
            sv = __builtin_amdgcn_wmma_f32_16x16x4_f32(false, aE[c], false, bX[c],
                                                       (short)0, sv, false, false);
        }

        // Issue next group's loads + prefetch now so the serial scan hides latency
        const int nextOff = (g + 1 < NGRP) ? GRP : 0;
        v2f bXn[4];
#pragma unroll
        for (int c = 0; c < 4; ++c) bXn[c] = *(const v2f*)(px + nextOff + 4 * c);
        const int pfOff = (g + 2 < NGRP) ? 2 * GRP : 0;
        __builtin_prefetch(px + pfOff, 0, 3);

        // 16-step inter-block state scan, redundantly in all lanes:
        //   s_{k+1} = A^16 s_k + S_local[:,k]
        // S_local[r][k] lives in lane k, accumulator component r (rows 0..3).
        float bs0 = 0.0f, bs1 = 0.0f;   // block-start state B-fragment for this lane
#pragma unroll
        for (int k = 0; k < 16; ++k) {
            if (k == m) {               // capture state at start of block m
                bs0 = half ? s2 : s0;
                bs1 = half ? s3 : s1;
            }
            float v0 = rdlane(sv[0], k);
            float v1 = rdlane(sv[1], k);
            float v2 = rdlane(sv[2], k);
            float v3 = rdlane(sv[3], k);
            float t0 = fmaf(A[0],  s0, fmaf(A[1],  s1, fmaf(A[2],  s2, fmaf(A[3],  s3, v0))));
            float t1 = fmaf(A[4],  s0, fmaf(A[5],  s1, fmaf(A[6],  s2, fmaf(A[7],  s3, v1))));
            float t2 = fmaf(A[8],  s0, fmaf(A[9],  s1, fmaf(A[10], s2, fmaf(A[11], s3, v2))));
            float t3 = fmaf(A[12], s0, fmaf(A[13], s1, fmaf(A[14], s2, fmaf(A[15], s3, v3))));
            s0 = t0; s1 = t1; s2 = t2; s3 = t3;
        }
        // s0..s3 now hold the next group's carry (uniform across lanes)

        // Correction: Y += H_pad @ S_pad (only K-chunk 0 nonzero -> 1 WMMA)
        v2f bS; bS[0] = bs0; bS[1] = bs1;
        yv = __builtin_amdgcn_wmma_f32_16x16x4_f32(false, aH, false, bS,
                                                   (short)0, yv, false, false);

        // Store: lane holds column (block) n=m, rows r+8*half -> 8 contiguous floats
        {
            float* po = yout + g * GRP + m * 16 + 8 * half;
            float4 o0 = { yv[0], yv[1], yv[2], yv[3] };
            float4 o1 = { yv[4], yv[5], yv[6], yv[7] };
            *(float4*)(po + 0) = o0;
            *(float4*)(po + 4) = o1;
        }

#pragma unroll
        for (int c = 0; c < 4; ++c) bX[c] = bXn[c];
        px += GRP;
    }
}

extern "C" void kernel_launch(void* const* d_in, const int* in_sizes, int n_in,
                              void* d_out, int out_size, void* d_ws, size_t ws_size,
                              hipStream_t stream)
{
    const float* x   = (const float*)d_in[0];
    const float* sos = (const float*)d_in[1];
    float* out = (float*)d_out;
    const int nsig = in_sizes[0] / T_LEN;   // 32*8 = 256 channels
    biquad_wmma_kernel<<<nsig, 32, 0, stream>>>(x, sos, out);
}